// GCN_59562606461344
// MI455X (gfx1250) — compile-verified
//
#include <hip/hip_runtime.h>
#include <hip/hip_bf16.h>

// GCN 2-layer: h = relu(Agg(X@W1)+b1); out = Agg(h@W2)+b2
// Agg = sym-normalized scatter-add with self loops (norm = dinv[src]*dinv[dst]).

typedef __attribute__((ext_vector_type(2))) float v2f;
typedef __attribute__((ext_vector_type(8))) float v8f;

#define NN  100000
#define EE  1600000
#define EMBD 128
#define HIDD 64
#define REPD 32

// ---------------- degree / dinv ----------------
__global__ void deg_init_k(float* __restrict__ deg) {
    int i = blockIdx.x * blockDim.x + threadIdx.x;
    if (i < NN) deg[i] = 1.0f;                 // self-loop
}

__global__ void deg_count_k(const int* __restrict__ dst, float* __restrict__ deg) {
    int e = blockIdx.x * blockDim.x + threadIdx.x;
    if (e < EE) atomicAdd(&deg[dst[e]], 1.0f);
}

__global__ void dinv_k(float* __restrict__ deg) {
    int i = blockIdx.x * blockDim.x + threadIdx.x;
    if (i < NN) {
        float d = deg[i];
        deg[i] = (d > 0.0f) ? rsqrtf(d) : 0.0f;
    }
}

// ---------------- GEMM1: H1[N,64] = X[N,128] @ W1[128,64] (fp32 WMMA) -------
// Block: 256 thr = 8 waves; wave = (mtile in 0..1) x (ntile in 0..3);
// block covers 32 rows x 64 cols. N % 32 == 0 so no bounds checks.
// W1 staged into LDS with gfx1250 async-to-LDS loads (ASYNCcnt).
__global__ void __launch_bounds__(256) gemm1_k(const float* __restrict__ X,
                                               const float* __restrict__ W1,
                                               float* __restrict__ H1) {
    __shared__ float sW[EMBD * HIDD];          // 32 KB
    const int tid = threadIdx.x;

    // Async stage W1 -> LDS: each thread issues 16B copies, 1024 floats/pass.
    for (int i = tid * 4; i < EMBD * HIDD; i += 256 * 4) {
        unsigned ldsoff = (unsigned)(uintptr_t)(&sW[i]);   // low 32b = LDS byte offset
        const float* g = W1 + i;
        asm volatile("global_load_async_to_lds_b128 %0, %1, off"
                     :: "v"(ldsoff), "v"(g) : "memory");
    }
    asm volatile("s_wait_asynccnt 0" ::: "memory");
    __syncthreads();

    const int lane  = tid & 31;
    const int wave  = tid >> 5;
    const int mtile = wave >> 2;               // 0..1
    const int ntile = wave & 3;                // 0..3
    const int mbase = blockIdx.x * 32 + mtile * 16;
    const int row   = mbase + (lane & 15);
    const int ncol  = ntile * 16 + (lane & 15);
    const int kb    = (lane >> 4) * 2;         // lanes 16-31 hold K+2,K+3

    const float* __restrict__ xrow = X + (size_t)row * EMBD;

    v8f acc = {};
#pragma unroll
    for (int k0 = 0; k0 < EMBD; k0 += 4) {
        v2f a, b;
        a[0] = xrow[k0 + kb];
        a[1] = xrow[k0 + kb + 1];
        b[0] = sW[(k0 + kb)     * HIDD + ncol];
        b[1] = sW[(k0 + kb + 1) * HIDD + ncol];
        acc = __builtin_amdgcn_wmma_f32_16x16x4_f32(false, a, false, b,
                                                    (short)0, acc, false, false);
    }

    const int half = (lane >> 4) * 8;          // C/D: lanes 16-31 hold M+8
#pragma unroll
    for (int r = 0; r < 8; ++r)
        H1[(size_t)(mbase + half + r) * HIDD + ncol] = acc[r];
}

// ---------------- self-loop init (float4 streaming writes) ------------------
template <int C, bool BIAS>
__global__ void selfloop_k(const float4* __restrict__ H, const float* __restrict__ dinv,
                           const float* __restrict__ bias, float4* __restrict__ agg) {
    long t = (long)blockIdx.x * blockDim.x + threadIdx.x;   // float4 index
    if (t >= (long)NN * (C / 4)) return;
    int i  = (int)(t / (C / 4));
    int c4 = (int)(t % (C / 4)) * 4;
    float di = dinv[i];
    float s  = di * di;
    float4 h = H[t];
    float4 v = make_float4(h.x * s, h.y * s, h.z * s, h.w * s);
    if (BIAS) {
        v.x += bias[c4];
        v.y += bias[c4 + 1];
        v.z += bias[c4 + 2];
        v.w += bias[c4 + 3];
    }
    agg[t] = v;
}

// ---------------- edge scatter: one wave per edge ---------------------------
// Edge metadata forced wave-uniform (readfirstlane) so src/dst/dinv loads go
// scalar; the vector path carries only the coalesced gather + f32 atomics.
template <int C>
__global__ void __launch_bounds__(256) scatter_k(const float* __restrict__ H,
                                                 const int* __restrict__ src,
                                                 const int* __restrict__ dst,
                                                 const float* __restrict__ dinv,
                                                 float* __restrict__ agg) {
    const int lane = threadIdx.x & 31;
    const int wave = threadIdx.x >> 5;
    int e = blockIdx.x * 8 + wave;             // one edge per wave
    if (e >= EE) return;                       // wave-uniform branch
    e = __builtin_amdgcn_readfirstlane(e);
    const int s = __builtin_amdgcn_readfirstlane(src[e]);
    const int d = __builtin_amdgcn_readfirstlane(dst[e]);
    const float norm = dinv[s] * dinv[d];
    const float* __restrict__ hrow = H   + (size_t)s * C;
    float* __restrict__       arow = agg + (size_t)d * C;
#pragma unroll
    for (int c = lane; c < C; c += 32)
        atomicAdd(&arow[c], hrow[c] * norm);
}

// ---------------- GEMM2: H2[N,32] = relu(agg1+b1)[N,64] @ W2[64,32] ---------
// Block: 256 thr = 8 waves; wave = (mtile 0..3) x (ntile 0..1); 64 rows/block.
__global__ void __launch_bounds__(256) gemm2_k(const float* __restrict__ A,
                                               const float* __restrict__ b1,
                                               const float* __restrict__ W2,
                                               float* __restrict__ H2) {
    __shared__ float sW[HIDD * REPD];          // 8 KB
    __shared__ float sB[HIDD];
    const int tid = threadIdx.x;
    for (int i = tid; i < HIDD * REPD; i += 256) sW[i] = W2[i];
    if (tid < HIDD) sB[tid] = b1[tid];
    __syncthreads();

    const int lane  = tid & 31;
    const int wave  = tid >> 5;
    const int mtile = wave >> 1;               // 0..3
    const int ntile = wave & 1;                // 0..1
    const int mbase = blockIdx.x * 64 + mtile * 16;
    int row = mbase + (lane & 15);
    if (row >= NN) row = NN - 1;               // clamp: loads stay in range
    const int ncol = ntile * 16 + (lane & 15);
    const int kb   = (lane >> 4) * 2;

    const float* __restrict__ arow = A + (size_t)row * HIDD;

    v8f acc = {};
#pragma unroll
    for (int k0 = 0; k0 < HIDD; k0 += 4) {
        v2f a, b;
        float a0 = arow[k0 + kb]     + sB[k0 + kb];
        float a1 = arow[k0 + kb + 1] + sB[k0 + kb + 1];
        a[0] = fmaxf(a0, 0.0f);                // fused bias + ReLU
        a[1] = fmaxf(a1, 0.0f);
        b[0] = sW[(k0 + kb)     * REPD + ncol];
        b[1] = sW[(k0 + kb + 1) * REPD + ncol];
        acc = __builtin_amdgcn_wmma_f32_16x16x4_f32(false, a, false, b,
                                                    (short)0, acc, false, false);
    }

    const int half = (lane >> 4) * 8;
#pragma unroll
    for (int r = 0; r < 8; ++r) {
        int m = mbase + half + r;
        if (m < NN)
            H2[(size_t)m * REPD + ncol] = acc[r];
    }
}

// ---------------------------------------------------------------------------
extern "C" void kernel_launch(void* const* d_in, const int* in_sizes, int n_in,
                              void* d_out, int out_size, void* d_ws, size_t ws_size,
                              hipStream_t stream) {
    const float* x    = (const float*)d_in[0];   // [N,128]
    const int*   edge = (const int*)d_in[1];     // [2,E]
    const float* W1   = (const float*)d_in[2];   // [128,64]
    const float* b1   = (const float*)d_in[3];   // [64]
    const float* W2   = (const float*)d_in[4];   // [64,32]
    const float* b2   = (const float*)d_in[5];   // [32]
    float* out = (float*)d_out;                  // [N,32]

    const int* src = edge;
    const int* dst = edge + EE;

    // workspace layout (floats)
    float* ws   = (float*)d_ws;
    float* deg  = ws;                                        // N (becomes dinv)
    float* H1   = ws + (((size_t)NN + 255) & ~(size_t)255);  // N*64
    float* agg1 = H1 + (size_t)NN * HIDD;                    // N*64
    float* H2   = agg1 + (size_t)NN * HIDD;                  // N*32

    const int T = 256;
    // 1) degrees -> dinv (in place)
    deg_init_k<<<(NN + T - 1) / T, T, 0, stream>>>(deg);
    deg_count_k<<<(EE + T - 1) / T, T, 0, stream>>>(dst, deg);
    dinv_k<<<(NN + T - 1) / T, T, 0, stream>>>(deg);

    // 2) layer 1 linear: H1 = X @ W1   (N % 32 == 0)
    gemm1_k<<<NN / 32, T, 0, stream>>>(x, W1, H1);

    // 3) layer 1 aggregation: agg1 = H1[i]*dinv^2 (self loop, init) + scatter
    {
        long q = (long)NN * (HIDD / 4);
        selfloop_k<HIDD, false><<<(int)((q + T - 1) / T), T, 0, stream>>>(
            (const float4*)H1, deg, nullptr, (float4*)agg1);
        scatter_k<HIDD><<<(EE + 7) / 8, T, 0, stream>>>(H1, src, dst, deg, agg1);
    }

    // 4) layer 2 linear (fused bias1+relu on A): H2 = relu(agg1+b1) @ W2
    gemm2_k<<<(NN + 63) / 64, T, 0, stream>>>(agg1, b1, W2, H2);

    // 5) layer 2 aggregation straight into d_out (init includes b2)
    {
        long q = (long)NN * (REPD / 4);
        selfloop_k<REPD, true><<<(int)((q + T - 1) / T), T, 0, stream>>>(
            (const float4*)H2, deg, b2, (float4*)out);
        scatter_k<REPD><<<(EE + 7) / 8, T, 0, stream>>>(H2, src, dst, deg, out);
    }
}